// ForwardAttention_8452495638852
// MI455X (gfx1250) — compile-verified
//
#include <hip/hip_runtime.h>
#include <hip/hip_bf16.h>
#include <math.h>

// ---------------------------------------------------------------------------
// Shapes (fixed by the reference):
//   B=128, T=2048, RNN=1024, EMB=512, ATTN=128, NF=32, KSZ=31
// Outputs (concatenated): context [128,512] then alpha_t [128,2048]
// ---------------------------------------------------------------------------
#define B_    128
#define T_    2048
#define RNN_  1024
#define EMB_  512
#define ATTN_ 128
#define NF_   32
#define KSZ_  31
#define PAD_  15

typedef __attribute__((ext_vector_type(2))) float v2f;
typedef __attribute__((ext_vector_type(8))) float v8f;

// ---------------------------------------------------------------------------
// Kernel A: pq[b,a] = sum_k hidden[b,k] * Wq[a,k]   (128x128 = 8x8 WMMA tiles)
// One wave32 per 16x16 tile; K looped in steps of 4 using V_WMMA_F32_16X16X4_F32.
// A-frag layout (ISA 7.12.2, 32-bit A 16x4): lanes 0-15 -> M rows, VGPR0=K0|K2,
// VGPR1=K1|K3 (low|high half-wave). B assumed symmetric (lanes->N, VGPRs->K).
// C layout: VGPR v -> M = v + 8*(lane>=16), lane&15 -> N.
// ---------------------------------------------------------------------------
__global__ __launch_bounds__(32)
void fa_query_wmma(const float* __restrict__ hidden,
                   const float* __restrict__ Wq,
                   float* __restrict__ pq) {
  const int tileM = blockIdx.x;          // b tile  0..7
  const int tileN = blockIdx.y;          // a tile  0..7
  const int lane  = threadIdx.x;         // 0..31
  const int half  = lane >> 4;           // 0 | 1
  const int l     = lane & 15;
  const int m     = tileM * 16 + l;      // batch row for A frag
  const int n     = tileN * 16 + l;      // attn col for B frag

  const float* __restrict__ arow = hidden + (size_t)m * RNN_;
  const float* __restrict__ brow = Wq     + (size_t)n * RNN_;   // B[k][n] = Wq[n][k]

  v8f c = {};
  for (int k = 0; k < RNN_; k += 4) {
    const int k0 = k + half * 2;
    v2f a, b;
    a.x = arow[k0];
    a.y = arow[k0 + 1];
    b.x = brow[k0];
    b.y = brow[k0 + 1];
    c = __builtin_amdgcn_wmma_f32_16x16x4_f32(
            /*neg_a=*/false, a, /*neg_b=*/false, b,
            /*c_mod=*/(short)0, c, /*reuse_a=*/false, /*reuse_b=*/false);
  }
#pragma unroll
  for (int v = 0; v < 8; ++v) {
    const int mm = tileM * 16 + v + half * 8;
    pq[(size_t)mm * ATTN_ + tileN * 16 + l] = c[v];
  }
}

// ---------------------------------------------------------------------------
// deterministic block reductions (fixed tree order)
// ---------------------------------------------------------------------------
__device__ __forceinline__ float block_max(float v, float* red, int tid) {
  red[tid] = v; __syncthreads();
  for (int s = 128; s > 0; s >>= 1) {
    if (tid < s) red[tid] = fmaxf(red[tid], red[tid + s]);
    __syncthreads();
  }
  float r = red[0]; __syncthreads();
  return r;
}
__device__ __forceinline__ float block_sum(float v, float* red, int tid) {
  red[tid] = v; __syncthreads();
  for (int s = 128; s > 0; s >>= 1) {
    if (tid < s) red[tid] += red[tid + s];
    __syncthreads();
  }
  float r = red[0]; __syncthreads();
  return r;
}

// ---------------------------------------------------------------------------
// Kernel B: one block (256 threads) per batch row b.
//   conv(awc) -> loc[32][t] ; e[t] = sum_a Wv[a]*tanh(pq[a]+pa[t,a]+pm[t,a])
//   softmax -> forward-attention recurrence -> softmax -> alpha_t (to d_out)
// processed_memory (128 MB) is streamed exactly once, 512B contiguous per t.
// ---------------------------------------------------------------------------
__global__ __launch_bounds__(256)
void fa_energies_softmax(const float* __restrict__ pm,
                         const float* __restrict__ awc,
                         const float* __restrict__ prev,
                         const unsigned char* __restrict__ mask,
                         const float* __restrict__ pq,
                         const float* __restrict__ convw,
                         const float* __restrict__ Wloc,
                         const float* __restrict__ Wv,
                         float* __restrict__ out_alpha) {
  const int b   = blockIdx.x;
  const int tid = threadIdx.x;

  __shared__ float awc_s[2][T_ + 32];       // halo 15 each side (index t+15)
  __shared__ float convw_s[NF_ * 2 * KSZ_]; // same layout as global (f,c,k)
  __shared__ float wloc_s[ATTN_ * NF_];
  __shared__ float pq_s[ATTN_];
  __shared__ float wv_s[ATTN_];
  __shared__ float e_s[T_];
  __shared__ float red_s[256];

  // ---- stage weights / activations into LDS --------------------------------
  for (int i = tid; i < 2 * (T_ + 32); i += 256) {
    const int c = i / (T_ + 32);
    const int t = (i % (T_ + 32)) - PAD_;
    awc_s[c][t + PAD_] = (t >= 0 && t < T_)
        ? awc[((size_t)b * 2 + c) * T_ + t] : 0.0f;
  }
  for (int i = tid; i < NF_ * 2 * KSZ_; i += 256) convw_s[i] = convw[i];
  for (int i = tid; i < ATTN_ * NF_;   i += 256) wloc_s[i]  = Wloc[i];
  if (tid < ATTN_) { pq_s[tid] = pq[(size_t)b * ATTN_ + tid]; wv_s[tid] = Wv[tid]; }
  __syncthreads();

  // ---- conv + energies -----------------------------------------------------
  for (int j = 0; j < T_ / 256; ++j) {
    const int t = j * 256 + tid;
    float locf[NF_];
#pragma unroll
    for (int f = 0; f < NF_; ++f) {
      const float* wf = &convw_s[f * 2 * KSZ_];
      float s = 0.0f;
#pragma unroll
      for (int k = 0; k < KSZ_; ++k) {
        s = fmaf(awc_s[0][t + k], wf[k], s);
        s = fmaf(awc_s[1][t + k], wf[KSZ_ + k], s);
      }
      locf[f] = s;
    }
    float e = 0.0f;
    const float* __restrict__ pmrow = pm + ((size_t)b * T_ + t) * ATTN_;
    for (int a = 0; a < ATTN_; ++a) {
      float dot = 0.0f;
#pragma unroll
      for (int f = 0; f < NF_; ++f) dot = fmaf(wloc_s[a * NF_ + f], locf[f], dot);
      const float s = pq_s[a] + pmrow[a] + dot;
      e = fmaf(wv_s[a], tanhf(s), e);
    }
    e_s[t] = mask[(size_t)b * T_ + t] ? -INFINITY : e;
  }
  __syncthreads();

  // ---- softmax #1 ----------------------------------------------------------
  float lmax = -INFINITY;
  for (int j = 0; j < T_ / 256; ++j) lmax = fmaxf(lmax, e_s[j * 256 + tid]);
  const float gmax = block_max(lmax, red_s, tid);

  float lsum = 0.0f;
  for (int j = 0; j < T_ / 256; ++j) {
    const int t = j * 256 + tid;
    const float ex = expf(e_s[t] - gmax);
    e_s[t] = ex;
    lsum += ex;
  }
  const float gsum = block_sum(lsum, red_s, tid);
  const float inv1 = 1.0f / gsum;

  // ---- forward-attention recurrence: alpha * shift(prev), renormalize ------
  float lsum2 = 0.0f;
  for (int j = 0; j < T_ / 256; ++j) {
    const int t = j * 256 + tid;
    const float pv = (t > 0) ? prev[(size_t)b * T_ + t - 1] : 0.0f;
    const float ap = (e_s[t] * inv1) * pv;
    e_s[t] = ap;
    lsum2 += ap;
  }
  const float gsum2 = block_sum(lsum2, red_s, tid);
  const float inv2  = 1.0f / (gsum2 + 1e-8f);

  // ---- softmax #2 over (alpha2 + 1e-8) ------------------------------------
  float lmax2 = -INFINITY;
  for (int j = 0; j < T_ / 256; ++j) {
    const int t = j * 256 + tid;
    const float x = e_s[t] * inv2 + 1e-8f;
    e_s[t] = x;
    lmax2 = fmaxf(lmax2, x);
  }
  const float gmax2 = block_max(lmax2, red_s, tid);

  float lsum3 = 0.0f;
  for (int j = 0; j < T_ / 256; ++j) {
    const int t = j * 256 + tid;
    const float ex = expf(e_s[t] - gmax2);
    e_s[t] = ex;
    lsum3 += ex;
  }
  const float gsum3 = block_sum(lsum3, red_s, tid);
  const float inv3  = 1.0f / gsum3;

  for (int j = 0; j < T_ / 256; ++j) {
    const int t = j * 256 + tid;
    out_alpha[(size_t)b * T_ + t] = e_s[t] * inv3;
  }
}

// ---------------------------------------------------------------------------
// Kernel C: context[b,d] = sum_t alpha[b,t] * memory[b,t,d]
// grid (4, 128): block owns 128 consecutive d's -> every 512B line of the
// 512 MB `memory` tensor is read exactly once, fully coalesced. alpha staged
// through LDS in 256-wide chunks (broadcast reads).
// ---------------------------------------------------------------------------
__global__ __launch_bounds__(128)
void fa_context(const float* __restrict__ memory,
                const float* __restrict__ alpha,
                float* __restrict__ ctx) {
  const int b = blockIdx.y;
  const int d = blockIdx.x * 128 + threadIdx.x;

  __shared__ float a_s[256];
  const float* __restrict__ mrow = memory + (size_t)b * T_ * EMB_;
  const float* __restrict__ arow = alpha + (size_t)b * T_;

  float acc = 0.0f;
  for (int tc = 0; tc < T_; tc += 256) {
    a_s[threadIdx.x]       = arow[tc + threadIdx.x];
    a_s[threadIdx.x + 128] = arow[tc + threadIdx.x + 128];
    __syncthreads();
#pragma unroll 4
    for (int j = 0; j < 256; ++j)
      acc = fmaf(a_s[j], mrow[(size_t)(tc + j) * EMB_ + d], acc);
    __syncthreads();
  }
  ctx[(size_t)b * EMB_ + d] = acc;
}

// ---------------------------------------------------------------------------
extern "C" void kernel_launch(void* const* d_in, const int* in_sizes, int n_in,
                              void* d_out, int out_size, void* d_ws, size_t ws_size,
                              hipStream_t stream) {
  (void)in_sizes; (void)n_in; (void)out_size; (void)ws_size;

  const float*         hidden = (const float*)d_in[0];         // [128,1024]
  const float*         memory = (const float*)d_in[1];         // [128,2048,512]
  const float*         pm     = (const float*)d_in[2];         // [128,2048,128]
  const float*         awc    = (const float*)d_in[3];         // [128,2,2048]
  const float*         prev   = (const float*)d_in[4];         // [128,2048]
  const unsigned char* mask   = (const unsigned char*)d_in[5]; // [128,2048] bool
  const float*         Wq     = (const float*)d_in[6];         // [128,1024]
  const float*         convw  = (const float*)d_in[7];         // [32,2,31]
  const float*         Wloc   = (const float*)d_in[8];         // [128,32]
  const float*         Wv     = (const float*)d_in[9];         // [1,128]

  float* out_ctx   = (float*)d_out;                 // [128,512]
  float* out_alpha = out_ctx + (size_t)B_ * EMB_;   // [128,2048]
  float* pq_ws     = (float*)d_ws;                  // [128,128] scratch

  // A: query projection via V_WMMA_F32_16X16X4_F32
  fa_query_wmma<<<dim3(8, 8), 32, 0, stream>>>(hidden, Wq, pq_ws);

  // B: fused conv + energies + masked softmax + forward-attn recurrence
  fa_energies_softmax<<<B_, 256, 0, stream>>>(pm, awc, prev, mask, pq_ws,
                                              convw, Wloc, Wv, out_alpha);

  // C: context GEMV, streams `memory` once fully coalesced
  fa_context<<<dim3(4, B_), 128, 0, stream>>>(memory, out_alpha, out_ctx);
}